// LiteCamProjector_82197084111485
// MI455X (gfx1250) — compile-verified
//
#include <hip/hip_runtime.h>
#include <stdint.h>

// ---------------- configuration ----------------
#define BLOCK  256                 // 8 wave32 waves
#define TILE   4096                // points per block tile
#define ROUNDS (TILE / BLOCK)      // 16
#define NWAVES (BLOCK / 32)        // 8
#define SCANB  512                 // threads in partial-scan kernel

// ranges (double, matching Python constants; float versions for elementwise ops)
#define XR0 (-51.2)
#define XR1 ( 51.2)
#define YR0 (-51.2)
#define YR1 ( 51.2)
#define ZR0 ( -5.0)
#define ZR1 (  3.0)
#define XR0F ((float)XR0)
#define XR1F ((float)XR1)
#define YR0F ((float)YR0)
#define YR1F ((float)YR1)
#define ZR0F ((float)ZR0)
#define ZR1F ((float)ZR1)

// ---------------- CDNA5 feature probes ----------------
#if defined(__has_builtin)
#  if __has_builtin(__builtin_amdgcn_ballot_w32)
#    define BALLOT32(p) __builtin_amdgcn_ballot_w32(p)
#  endif
#  if __has_builtin(__builtin_amdgcn_global_load_async_to_lds_b128)
#    define HAVE_ASYNC_LDS 1
#  endif
#endif
#ifndef BALLOT32
#  define BALLOT32(p) ((unsigned)__ballot(p))
#endif

__device__ __forceinline__ void async_wait_all() {
#if defined(__has_builtin) && __has_builtin(__builtin_amdgcn_s_wait_asynccnt)
    __builtin_amdgcn_s_wait_asynccnt(0);
#else
    asm volatile("s_wait_asynccnt 0" ::: "memory");
#endif
}

// 128-bit vector type matching the async-LDS builtin's pointee type
typedef int v4i_cd5 __attribute__((ext_vector_type(4)));
typedef __attribute__((address_space(1))) v4i_cd5* gbl_v4i_p;   // global (AS1)
typedef __attribute__((address_space(3))) v4i_cd5* lds_v4i_p;   // LDS    (AS3)

// 16-byte global -> LDS copy (async on CDNA5, sync fallback otherwise)
__device__ __forceinline__ void cp16_async(void* lds_dst, const void* g_src) {
#if defined(HAVE_ASYNC_LDS)
    __builtin_amdgcn_global_load_async_to_lds_b128(
        (gbl_v4i_p)g_src,
        (lds_v4i_p)lds_dst,
        0, 0);
#else
    *(float4*)lds_dst = *(const float4*)g_src;
#endif
}

// ---------------- projection parameters ----------------
struct Proj {
    float fx, fy, cx, cy;
    float r00, r01, r02, r10, r11, r12, r20, r21, r22;
    float tx, ty, tz;
    float Wm1, Hm1;          // W-1, H-1 for clipping
    float dxf, dyf;          // bin sizes (computed in double, as reference)
    float WbM1, HbM1;        // Wb-1, Hb-1
};

__device__ __forceinline__ void load_params(const float* Km, const float* Tm,
                                            const int* pH, const int* pW,
                                            const int* pHb, const int* pWb,
                                            Proj& P) {
    P.fx = Km[0]; P.fy = Km[4]; P.cx = Km[2]; P.cy = Km[5];
    P.r00 = Tm[0]; P.r01 = Tm[1]; P.r02 = Tm[2];  P.tx = Tm[3];
    P.r10 = Tm[4]; P.r11 = Tm[5]; P.r12 = Tm[6];  P.ty = Tm[7];
    P.r20 = Tm[8]; P.r21 = Tm[9]; P.r22 = Tm[10]; P.tz = Tm[11];
    int H = *pH, W = *pW, Hb = *pHb, Wb = *pWb;
    P.Wm1  = (float)(W - 1);
    P.Hm1  = (float)(H - 1);
    P.dxf  = (float)((XR1 - XR0) / (double)Wb);  // matches Python double then f32
    P.dyf  = (float)((YR1 - YR0) / (double)Hb);
    P.WbM1 = (float)(Wb - 1);
    P.HbM1 = (float)(Hb - 1);
}

// exact elementwise op order of the reference, contraction off
__device__ __forceinline__ bool project(const Proj& P, float pu, float pv, float d,
                                        int& oi, int& oj) {
#pragma clang fp contract(off)
    float u  = fminf(fmaxf(pu, 0.0f), P.Wm1);
    float v  = fminf(fmaxf(pv, 0.0f), P.Hm1);
    float Xc = (u - P.cx) / P.fx * d;
    float Yc = (v - P.cy) / P.fy * d;
    float Zc = d;
    float x = P.r00 * Xc + P.r01 * Yc + P.r02 * Zc + P.tx;
    float y = P.r10 * Xc + P.r11 * Yc + P.r12 * Zc + P.ty;
    float z = P.r20 * Xc + P.r21 * Yc + P.r22 * Zc + P.tz;
    bool m = (x >= XR0F) && (x < XR1F) &&
             (y >= YR0F) && (y < YR1F) &&
             (z >= ZR0F) && (z < ZR1F);
    float fj = floorf((x - XR0F) / P.dxf);
    fj = fminf(fmaxf(fj, 0.0f), P.WbM1);
    float fi = floorf((y - YR0F) / P.dyf);
    fi = fminf(fmaxf(fi, 0.0f), P.HbM1);
    oj = (int)fj;
    oi = (int)fi;
    return m;
}

// ---------------- tile staging (global -> LDS) ----------------
template <bool FULL>
__device__ __forceinline__ void stage_tile(const float2* pix, const float* depth,
                                           int tileStart, int N,
                                           float* sDepth, float2* sPix) {
    const int t = threadIdx.x;
    if (FULL) {
        const char* dg = (const char*)(depth + tileStart);
        char*       dl = (char*)sDepth;
        for (int c = t; c < (TILE * 4) / 16; c += BLOCK)   // 16 KB depth
            cp16_async(dl + c * 16, dg + c * 16);
        const char* pg = (const char*)(pix + tileStart);
        char*       pl = (char*)sPix;
        for (int c = t; c < (TILE * 8) / 16; c += BLOCK)   // 32 KB pix_uv
            cp16_async(pl + c * 16, pg + c * 16);
        async_wait_all();
    } else {
        for (int e = t; e < TILE; e += BLOCK) {
            int g = tileStart + e;
            if (g < N) { sDepth[e] = depth[g]; sPix[e] = pix[g]; }
            else       { sDepth[e] = 0.0f;     sPix[e] = make_float2(0.0f, 0.0f); }
        }
    }
    __syncthreads();
}

// ---------------- kernel 1: mask + zero-fill + per-tile counts ----------------
__global__ void __launch_bounds__(BLOCK)
k_mask_count(const float2* pix, const float* depth,
             const float* Km, const float* Tm,
             const int* pH, const int* pW, const int* pHb, const int* pWb,
             float* out, unsigned* partials, int N) {
    __shared__ float  sDepth[TILE];
    __shared__ float2 sPix[TILE];
    __shared__ unsigned sWave[NWAVES];

    Proj P;
    load_params(Km, Tm, pH, pW, pHb, pWb, P);

    const int tile = blockIdx.x;
    const int tileStart = tile * TILE;
    const int t = threadIdx.x;
    const int lane = t & 31;
    const int w = t >> 5;

    __builtin_prefetch(depth + tileStart, 0, 1);   // global_prefetch_b8
    __builtin_prefetch(pix + tileStart, 0, 1);

    if (tileStart + TILE <= N) stage_tile<true >(pix, depth, tileStart, N, sDepth, sPix);
    else                       stage_tile<false>(pix, depth, tileStart, N, sDepth, sPix);

    unsigned waveTotal = 0;
    for (int r = 0; r < ROUNDS; ++r) {
        const int e = r * BLOCK + t;
        const int g = tileStart + e;
        int ii, jj;
        bool m = project(P, sPix[e].x, sPix[e].y, sDepth[e], ii, jj) && (g < N);
        unsigned b = BALLOT32(m);
        waveTotal += (unsigned)__popc(b);
        if (g < N) {
            out[g] = m ? 1.0f : 0.0f;                      // mask output
            out[(size_t)N + 2 * (size_t)g]     = 0.0f;     // zero ij_out
            out[(size_t)N + 2 * (size_t)g + 1] = 0.0f;
        }
    }
    if (lane == 0) sWave[w] = waveTotal;
    __syncthreads();
    if (t == 0) {
        unsigned tot = 0;
        for (int q = 0; q < NWAVES; ++q) tot += sWave[q];
        partials[tile] = tot;
    }
}

// ---------------- kernel 2: exclusive scan of tile partials + count ----------------
__global__ void __launch_bounds__(SCANB)
k_scan_partials(const unsigned* partials, unsigned* offsets, float* out,
                int numTiles, int N) {
    __shared__ unsigned s[SCANB];
    __shared__ unsigned sCarry;
    const int t = threadIdx.x;
    if (t == 0) sCarry = 0;
    __syncthreads();

    for (int base = 0; base < numTiles; base += SCANB) {
        const int idx = base + t;
        unsigned v = (idx < numTiles) ? partials[idx] : 0u;
        s[t] = v;
        __syncthreads();
        for (int off = 1; off < SCANB; off <<= 1) {
            unsigned u = (t >= off) ? s[t - off] : 0u;
            __syncthreads();
            s[t] += u;
            __syncthreads();
        }
        unsigned carry = sCarry;
        if (idx < numTiles) offsets[idx] = carry + s[t] - v;   // exclusive
        __syncthreads();
        if (t == SCANB - 1) sCarry = carry + s[SCANB - 1];
        __syncthreads();
    }
    if (t == 0) out[3 * (size_t)N] = (float)sCarry;            // count
}

// ---------------- kernel 3: rescan + ordered scatter ----------------
__global__ void __launch_bounds__(BLOCK)
k_scatter(const float2* pix, const float* depth,
          const float* Km, const float* Tm,
          const int* pH, const int* pW, const int* pHb, const int* pWb,
          const unsigned* offsets, float* out, int N) {
    __shared__ float  sDepth[TILE];
    __shared__ float2 sPix[TILE];
    __shared__ unsigned sW[NWAVES];

    Proj P;
    load_params(Km, Tm, pH, pW, pHb, pWb, P);

    const int tile = blockIdx.x;
    const int tileStart = tile * TILE;
    const int t = threadIdx.x;
    const int lane = t & 31;
    const int w = t >> 5;
    const unsigned ltmask = (1u << lane) - 1u;

    __builtin_prefetch(depth + tileStart, 0, 1);
    __builtin_prefetch(pix + tileStart, 0, 1);

    if (tileStart + TILE <= N) stage_tile<true >(pix, depth, tileStart, N, sDepth, sPix);
    else                       stage_tile<false>(pix, depth, tileStart, N, sDepth, sPix);

    unsigned running = offsets[tile];
    float* ij = out + (size_t)N;

    for (int r = 0; r < ROUNDS; ++r) {
        const int e = r * BLOCK + t;
        const int g = tileStart + e;
        int ii, jj;
        bool m = project(P, sPix[e].x, sPix[e].y, sDepth[e], ii, jj) && (g < N);
        unsigned b = BALLOT32(m);
        if (lane == 0) sW[w] = (unsigned)__popc(b);
        __syncthreads();
        unsigned waveOff = 0, roundTot = 0;
        for (int q = 0; q < NWAVES; ++q) {
            unsigned c = sW[q];
            roundTot += c;
            if (q < w) waveOff += c;
        }
        if (m) {
            unsigned pos = running + waveOff + (unsigned)__popc(b & ltmask);
            ij[2 * (size_t)pos]     = (float)ii;
            ij[2 * (size_t)pos + 1] = (float)jj;
        }
        running += roundTot;
        __syncthreads();   // sW reused next round
    }
}

// ---------------- host launcher ----------------
extern "C" void kernel_launch(void* const* d_in, const int* in_sizes, int n_in,
                              void* d_out, int out_size, void* d_ws, size_t ws_size,
                              hipStream_t stream) {
    const float2* pix   = (const float2*)d_in[0];  // (N,2) f32
    const float*  depth = (const float*)d_in[1];   // (N,)  f32
    const float*  Km    = (const float*)d_in[2];   // (3,3) f32
    const float*  Tm    = (const float*)d_in[3];   // (4,4) f32
    const int*    pH    = (const int*)d_in[4];
    const int*    pW    = (const int*)d_in[5];
    const int*    pHb   = (const int*)d_in[6];
    const int*    pWb   = (const int*)d_in[7];

    const int N = in_sizes[1];
    float* out = (float*)d_out;

    const int numTiles = (N + TILE - 1) / TILE;
    unsigned* partials = (unsigned*)d_ws;
    unsigned* offsets  = partials + numTiles;

    k_mask_count<<<numTiles, BLOCK, 0, stream>>>(pix, depth, Km, Tm, pH, pW, pHb, pWb,
                                                 out, partials, N);
    k_scan_partials<<<1, SCANB, 0, stream>>>(partials, offsets, out, numTiles, N);
    k_scatter<<<numTiles, BLOCK, 0, stream>>>(pix, depth, Km, Tm, pH, pW, pHb, pWb,
                                              offsets, out, N);
}